// NormalizedLocalMapping_77068893159604
// MI455X (gfx1250) — compile-verified
//
#include <hip/hip_runtime.h>

typedef __attribute__((ext_vector_type(2))) float v2f;
typedef __attribute__((ext_vector_type(8))) float v8f;
typedef __attribute__((ext_vector_type(4))) unsigned int u32x4;
typedef __attribute__((ext_vector_type(8))) int i32x8;
typedef __attribute__((ext_vector_type(4))) int i32x4;

#if defined(__has_builtin)
#if __has_builtin(__builtin_amdgcn_tensor_load_to_lds) && \
    __has_builtin(__builtin_amdgcn_s_wait_tensorcnt)
#define USE_TDM 1
#endif
#endif
#ifndef USE_TDM
#define USE_TDM 0
#endif

// Problem constants (match reference)
#define CIN      64
#define COUT     128
#define KFEAT    576          // CIN * 9
#define NCHUNK   8            // K chunks of 8 channels
#define KB       72           // 8 channels * 9 taps per chunk
#define AS_STRIDE 73          // 72 + 1 pad  (bank-conflict-free A-frag reads)
#define WS_STRIDE 136         // 128 + 8 pad (half-wave K-offset rows in disjoint banks)
#define AS_FLOATS (128 * AS_STRIDE)   // 9344
#define WS_FLOATS (KB * WS_STRIDE)    // 9792

// Double-buffered LDS layout (float offsets): As0 | As1 | Ws0 | Ws1
#define AS0_OFF   0
#define AS1_OFF   AS_FLOATS
#define WS0_OFF   (2 * AS_FLOATS)
#define WS1_OFF   (2 * AS_FLOATS + WS_FLOATS)
#define LDS_FLOATS (2 * AS_FLOATS + 2 * WS_FLOATS)   // 38272 floats = 153088 B

#define WTILE_ELTS 9216       // 72 * 128 contiguous floats per weight tile

__global__ __launch_bounds__(256)
void localconv_pool_wmma_f32(const float* __restrict__ x,      // (128,64,32,32)
                             const float* __restrict__ weight, // (1024,576,128)
                             const float* __restrict__ bias,   // (1024,128)
                             float* __restrict__ out) {        // (128,128,16,16)
    extern __shared__ float smem[];

    const int tid   = threadIdx.x;
    const int lane  = tid & 31;
    const int wv    = tid >> 5;        // 8 waves
    const int lo    = lane & 15;
    const int hi    = lane >> 4;       // half-wave select
    const int mbase = wv << 4;         // batch rows [mbase, mbase+16)

    const int pg = blockIdx.x;         // 256 pool groups
    const int ph = pg >> 4;
    const int pw = pg & 15;

    // ---- stage patch tile (unfold gather, L2-resident x): 128x72 -> As[buf]
    auto stageA = [&](int aoff, int lp, int cb) {
        const int oh = (ph << 1) + (lp >> 1);
        const int ow = (pw << 1) + (lp & 1);
        const int c0 = cb << 3;
        float* As = smem + aoff;
        for (int i = tid; i < 128 * KB; i += 256) {
            const int f = i % KB;
            const int b = i / KB;
            const int c   = c0 + f / 9;
            const int tap = f % 9;
            const int ih  = oh - 1 + tap / 3;
            const int iw  = ow - 1 + (tap % 3);
            float v = 0.0f;
            if ((unsigned)ih < 32u && (unsigned)iw < 32u)
                v = x[(((b * CIN + c) << 5) + ih) * 32 + iw];
            As[b * AS_STRIDE + f] = v;
        }
    };

    // ---- issue weight tile load: 72x128 contiguous -> Ws[buf] (stride 136)
    auto issueW = [&](int buf, int lp, int cb) {
        const int oh = (ph << 1) + (lp >> 1);
        const int ow = (pw << 1) + (lp & 1);
        const int l  = (oh << 5) + ow;
        const float* wsrc =
            weight + ((size_t)l * KFEAT + (size_t)(cb << 3) * 9) * COUT;
#if USE_TDM
        if (wv == 0) {
            // Tensor Data Mover: 1-D tile of 9216 f32, LDS padding of 8 DWORDs
            // after every 128 DWORDs reproduces the 136-float row stride.
            const unsigned lds_byte =
                (unsigned)((buf ? WS1_OFF : WS0_OFF) * sizeof(float));
            const unsigned long long ga = (unsigned long long)wsrc;
            u32x4 g0;
            g0.x = 1u;                                   // count=1 (valid D#)
            g0.y = lds_byte;                             // lds_addr (dyn LDS @0)
            g0.z = (unsigned)ga;                         // global_addr[31:0]
            g0.w = (unsigned)((ga >> 32) & 0x1FFFFFFull) // global_addr[56:32]
                   | (2u << 30);                         // type=2 ("image")
            i32x8 g1;
            g1[0] = (int)((2u << 16)      // data_size = 4 bytes
                        | (1u << 20)      // pad_enable
                        | (6u << 22)      // pad_interval: every 128 DWORDs
                        | (7u << 25));    // pad_amount: 8 DWORDs
            g1[1] = (int)((unsigned)WTILE_ELTS << 16); // tensor_dim0 lo16 @bit48
            g1[2] = (int)(1u << 16);                   // dim0 hi16=0, tensor_dim1=1
            g1[3] = (int)((unsigned)WTILE_ELTS << 16); // tile_dim0 = 9216
            g1[4] = 0;                                 // tile_dim1/2 unused
            g1[5] = (int)WTILE_ELTS;                   // tensor_dim0_stride
            g1[6] = 0;
            g1[7] = 0;
            i32x4 g2 = {0, 0, 0, 0};                   // group2 unused (<=2D)
            i32x4 g3 = {0, 0, 0, 0};                   // group3 unused (<=2D)
            i32x8 g4 = {0, 0, 0, 0, 0, 0, 0, 0};       // extra group (clang-23 form)
            __builtin_amdgcn_tensor_load_to_lds(g0, g1, g2, g3, g4, 0);
        }
#else
        float* Ws = smem + (buf ? WS1_OFF : WS0_OFF);
        for (int i = tid; i < KB * 32; i += 256) {       // 2304 float4 loads
            const float4 v = *(const float4*)(wsrc + (i << 2));
            const int f = i >> 5, g = i & 31;
            *(float4*)(Ws + f * WS_STRIDE + (g << 2)) = v;
        }
        __builtin_prefetch(wsrc + WTILE_ELTS + tid * 36, 0, 0);
#endif
    };

    v8f rmax[8] = {};                  // running pooled max (ReLU folded: >= 0)
    v8f acc[8];

    // Prologue: stage buffers for idx = 0
    issueW(0, 0, 0);
    stageA(AS0_OFF, 0, 0);

    // 32 iterations = 4 pool locations x 8 K-chunks, double-buffered
    for (int idx = 0; idx < 32; ++idx) {
        const int cur = idx & 1;
        const int lp  = idx >> 3;
        const int cb  = idx & 7;

#if USE_TDM
        if (wv == 0) __builtin_amdgcn_s_wait_tensorcnt(0);  // Ws[cur] landed
#endif
        __syncthreads();   // all staged data for [cur] visible; prev reads done

        // Kick next tile's loads before computing (overlap HBM stream w/ WMMA)
        if (idx + 1 < 32) {
            const int nlp = (idx + 1) >> 3;
            const int ncb = (idx + 1) & 7;
            issueW(1 - cur, nlp, ncb);
            stageA(cur ? AS0_OFF : AS1_OFF, nlp, ncb);
        }

        if (cb == 0) {
            v8f z = {};
#pragma unroll
            for (int nt = 0; nt < 8; ++nt) acc[nt] = z;
        }

        // ---- compute: 18 K-steps x 8 N-tiles of V_WMMA_F32_16X16X4_F32
        // A frag: M = lo, K = k + 2*hi + v ; B frag: K = k + 2*hi + v, N = nt*16+lo
        const float* As = smem + (cur ? AS1_OFF : AS0_OFF);
        const float* Ws = smem + (cur ? WS1_OFF : WS0_OFF);
        const float* Abase = As + (mbase + lo) * AS_STRIDE + (hi << 1);
        const float* Wbase = Ws + (hi << 1) * WS_STRIDE + lo;
        for (int kt = 0; kt < KB / 4; ++kt) {
            const float* Ap = Abase + (kt << 2);
            v2f a;
            a.x = Ap[0];
            a.y = Ap[1];
            const float* Wp = Wbase + (kt << 2) * WS_STRIDE;
#pragma unroll
            for (int nt = 0; nt < 8; ++nt) {
                v2f b;
                b.x = Wp[nt * 16];
                b.y = Wp[nt * 16 + WS_STRIDE];
                acc[nt] = __builtin_amdgcn_wmma_f32_16x16x4_f32(
                    false, a, false, b, (short)0, acc[nt], false, false);
            }
        }

        // ---- end of a location's K loop: bias + ReLU + pool-max into rmax
        if (cb == 7) {
            const int oh = (ph << 1) + (lp >> 1);
            const int ow = (pw << 1) + (lp & 1);
            const int l  = (oh << 5) + ow;
            const float* bptr = bias + l * COUT;
#pragma unroll
            for (int nt = 0; nt < 8; ++nt) {
                const float bv = bptr[nt * 16 + lo];
#pragma unroll
                for (int r = 0; r < 8; ++r)
                    rmax[nt][r] = fmaxf(rmax[nt][r], acc[nt][r] + bv);
            }
        }
    }

    // ---- write pooled output out[b][co][ph][pw], (128,128,16,16)
    // D layout: element r of lane -> (M = mbase + r + 8*hi, N = nt*16 + lo)
#pragma unroll
    for (int nt = 0; nt < 8; ++nt) {
        const int co = nt * 16 + lo;
#pragma unroll
        for (int r = 0; r < 8; ++r) {
            const int b = mbase + r + (hi << 3);
            out[(((b * COUT) + co) << 8) + (ph << 4) + pw] = rmax[nt][r];
        }
    }
}

extern "C" void kernel_launch(void* const* d_in, const int* in_sizes, int n_in,
                              void* d_out, int out_size, void* d_ws, size_t ws_size,
                              hipStream_t stream) {
    const float* x      = (const float*)d_in[0];
    const float* weight = (const float*)d_in[1];
    const float* bias   = (const float*)d_in[2];
    float* out = (float*)d_out;

    const size_t shmem = (size_t)LDS_FLOATS * sizeof(float); // 153088 B
    localconv_pool_wmma_f32<<<256, 256, shmem, stream>>>(x, weight, bias, out);
}